// RNNBase_30777735643461
// MI455X (gfx1250) — compile-verified
//
#include <hip/hip_runtime.h>

// Persistent-kernel LSTM (Keras gates i,f,g,o), B=64, T=512, D=512, H=512.
// Single launch. 128 blocks = 4 independent batch-tile groups x 32 column
// blocks. Each group runs the full 512-step recurrence, syncing per step via
// a sense-reversing L2 barrier (device-scope atomics).
//
// Per step per block:  z = [x_t ; h_{t-1}] @ [W ; R] + bias  via fp32 WMMA
// (v_wmma_f32_16x16x4_f32). W/R fragments are preloaded into VGPRs once and
// reused all 512 steps; cell state c, own h element, and prev_out stay in
// registers; only h is exchanged through global memory (ping-pong in d_ws).

typedef __attribute__((ext_vector_type(2))) float v2f;
typedef __attribute__((ext_vector_type(4))) float f4;
typedef __attribute__((ext_vector_type(8))) float v8f;

#define BB 64
#define TT 512
#define DD 512
#define HH 512
#define G4 2048          // 4*H
#define LDA 516          // padded LDS stride (floats): 516 % 64 == 4 -> conflict-free
#define NBLK 32          // blocks per batch-tile group

__global__ __launch_bounds__(512)
void lstm_persistent_kernel(const float* __restrict__ x,       // [B,T,D]
                            const float* __restrict__ Wk,      // [D,4H]
                            const float* __restrict__ Rk,      // [H,4H]
                            const float* __restrict__ bias,    // [4H]
                            const unsigned char* __restrict__ mask, // [B,T]
                            const float* __restrict__ init_h,  // [B,H]
                            const float* __restrict__ init_c,  // [B,H]
                            float* __restrict__ h0,            // [B,H] ping
                            float* __restrict__ h1,            // [B,H] pong
                            int*   __restrict__ bar,           // [4 groups][2]
                            float* __restrict__ out,           // [B,T,H]
                            float* __restrict__ hf,            // [B,H]
                            float* __restrict__ cf)            // [B,H]
{
    __shared__ float lds_a[16 * LDA];   // staged A rows (16 batch rows x 512 cols)
    __shared__ float lds_z[16 * 256];   // per-wave 16x16 fp32 partial tiles

    const int tid   = threadIdx.x;
    const int lane  = tid & 31;
    const int w     = tid >> 5;        // wave 0..15
    const int gi    = w & 3;           // gate index (i,f,g,o)
    const int kpart = w >> 2;          // 0..3: K partition within a 512-wide phase
    const int n0    = blockIdx.x * 16; // hidden-column tile
    const int m0    = blockIdx.y * 16; // batch-row tile (group id = blockIdx.y)

    // WMMA fp32 16x16x4 fragment addressing (wave32):
    //   A (16x4): lanes 0-15 -> M=lane, K={0,1}; lanes 16-31 -> M=lane-16, K={2,3}
    //   B (4x16): lanes 0-15 -> N=lane, K={0,1}; lanes 16-31 -> N=lane-16, K={2,3}
    const int mrow  = lane & 15;
    const int khalf = (lane >> 4) << 1;
    const int ncol  = lane & 15;
    const long gcol = (long)gi * HH + n0 + ncol;

    // ---- Preload this wave's W and R fragments into registers (held all run).
    v2f wfrag[32], rfrag[32];
    #pragma unroll
    for (int kk = 0; kk < 32; ++kk) {
        const int k = kpart * 128 + (kk << 2) + khalf;
        wfrag[kk].x = Wk[(long)k       * G4 + gcol];
        wfrag[kk].y = Wk[(long)(k + 1) * G4 + gcol];
        rfrag[kk].x = Rk[(long)k       * G4 + gcol];
        rfrag[kk].y = Rk[(long)(k + 1) * G4 + gcol];
    }

    // ---- Per-thread persistent element state (gate threads: tid < 256).
    const bool active = tid < 256;
    const int  row = (tid >> 4) & 15;     // batch row within tile
    const int  col = tid & 15;            // hidden col within tile
    const int  rp  = row & 7;             // partial-tile vgpr index
    const int  lp  = col + ((row & 8) ? 16 : 0);
    const int  m   = m0 + row;
    const int  n   = n0 + col;

    float creg = 0.0f, hreg = 0.0f, outprev = 0.0f;
    float bias_r[4] = {0.f, 0.f, 0.f, 0.f};
    if (active) {
        creg = init_c[m * HH + n];
        hreg = init_h[m * HH + n];
        #pragma unroll
        for (int g = 0; g < 4; ++g)
            bias_r[g] = bias[(long)g * HH + n0 + col];
    }

    int* cnt   = bar + blockIdx.y * 2;
    int* sense = bar + blockIdx.y * 2 + 1;
    int  lsense = 0;

    for (int t = 0; t < TT; ++t) {
        const float* hcur = (t & 1) ? h1 : h0;
        float*       hnxt = (t & 1) ? h0 : h1;

        v8f acc = {};

        // phase 0: A = x_t rows, B = W-frags ; phase 1: A = h_{t-1}, B = R-frags.
        #pragma unroll
        for (int phase = 0; phase < 2; ++phase) {
            __syncthreads();   // previous use of lds_a done
            #pragma unroll
            for (int i = 0; i < 4; ++i) {
                int id  = tid + i * 512;          // 2048 float4 slots
                int r   = id >> 7;                // 128 float4 per row
                int c4  = (id & 127) << 2;
                const float* src = (phase == 0)
                    ? (x    + ((long)(m0 + r) * TT + t) * DD + c4)
                    : (hcur +  (long)(m0 + r) * HH + c4);
                *(f4*)&lds_a[r * LDA + c4] = *(const f4*)src;
            }
            __syncthreads();

            const int kb = kpart * 128;
            #pragma unroll
            for (int kk = 0; kk < 32; ++kk) {
                const int k = kb + (kk << 2) + khalf;
                v2f a;
                a.x = lds_a[mrow * LDA + k];
                a.y = lds_a[mrow * LDA + k + 1];
                const v2f b = (phase == 0) ? wfrag[kk] : rfrag[kk];
                acc = __builtin_amdgcn_wmma_f32_16x16x4_f32(
                          false, a, false, b, (short)0, acc, false, false);
            }
        }

        // Reduce the 4 K-partials per gate through LDS.
        #pragma unroll
        for (int r = 0; r < 8; ++r)
            lds_z[w * 256 + r * 32 + lane] = acc[r];
        __syncthreads();

        if (active) {
            float z[4];
            #pragma unroll
            for (int g = 0; g < 4; ++g) {
                float s = bias_r[g];
                #pragma unroll
                for (int kp = 0; kp < 4; ++kp)
                    s += lds_z[(kp * 4 + g) * 256 + rp * 32 + lp];
                z[g] = s;
            }
            const float ig = 1.0f / (1.0f + __expf(-z[0]));
            const float fg = 1.0f / (1.0f + __expf(-z[1]));
            const float gg = tanhf(z[2]);
            const float og = 1.0f / (1.0f + __expf(-z[3]));

            const float cn = fg * creg + ig * gg;
            const float hn = og * tanhf(cn);
            const bool  mk = mask[(long)m * TT + t] != 0;

            hreg    = mk ? hn : hreg;
            creg    = mk ? cn : creg;
            outprev = mk ? hn : outprev;

            out[((long)m * TT + t) * HH + n] = outprev;
            hnxt[m * HH + n] = hreg;
            if (t == TT - 1) {
                hf[m * HH + n] = hreg;
                cf[m * HH + n] = creg;
            }
        }

        // ---- Group barrier: make hnxt visible to all 32 blocks of this group.
        __threadfence();           // release own stores to device scope
        __syncthreads();
        if (tid == 0) {
            const int s = lsense ^ 1;
            if (__hip_atomic_fetch_add(cnt, 1, __ATOMIC_ACQ_REL,
                                       __HIP_MEMORY_SCOPE_AGENT) == NBLK - 1) {
                __hip_atomic_store(cnt, 0, __ATOMIC_RELAXED, __HIP_MEMORY_SCOPE_AGENT);
                __hip_atomic_store(sense, s, __ATOMIC_RELEASE, __HIP_MEMORY_SCOPE_AGENT);
            } else {
                while (__hip_atomic_load(sense, __ATOMIC_ACQUIRE,
                                         __HIP_MEMORY_SCOPE_AGENT) != s)
                    __builtin_amdgcn_s_sleep(1);
            }
        }
        lsense ^= 1;
        __syncthreads();
        __threadfence();           // acquire: invalidate stale cached h lines
    }
}

extern "C" void kernel_launch(void* const* d_in, const int* in_sizes, int n_in,
                              void* d_out, int out_size, void* d_ws, size_t ws_size,
                              hipStream_t stream) {
    (void)in_sizes; (void)n_in; (void)out_size; (void)ws_size;

    const float*         x    = (const float*)d_in[0];          // inputs [B,T,D]
    const float*         ih   = (const float*)d_in[1];          // init_h [B,H]
    const float*         ic   = (const float*)d_in[2];          // init_c [B,H]
    const float*         Wk   = (const float*)d_in[3];          // kernel [D,4H]
    const float*         Rk   = (const float*)d_in[4];          // recurrent_kernel [H,4H]
    const float*         bias = (const float*)d_in[5];          // bias [4H]
    const unsigned char* mask = (const unsigned char*)d_in[6];  // mask [B,T] bool

    float* outp = (float*)d_out;                       // outputs [B,T,H]
    float* hf   = outp + (long)BB * TT * HH;           // final h [B,H]
    float* cf   = hf + (long)BB * HH;                  // final c [B,H]

    float* ws = (float*)d_ws;
    float* h0 = ws;                        // B*H floats
    float* h1 = ws + (long)BB * HH;        // B*H floats
    int*   bar = (int*)(ws + 2L * BB * HH);  // 4 groups x {count, sense}

    hipMemcpyAsync(h0, ih, sizeof(float) * BB * HH, hipMemcpyDeviceToDevice, stream);
    hipMemsetAsync(bar, 0, sizeof(int) * 8, stream);

    dim3 grid(HH / 16, BB / 16);   // 32 x 4 = 128 persistent blocks
    lstm_persistent_kernel<<<grid, 512, 0, stream>>>(
        x, Wk, Rk, bias, mask, ih, ic, h0, h1, bar, outp, hf, cf);
}